// LorentzSelfAttention_35304631173200
// MI455X (gfx1250) — compile-verified
//
#include <hip/hip_runtime.h>
#include <math.h>

// ---- problem constants (match reference) ----
#define Bv   2
#define Tv   512
#define Dm   512
#define Hh   8
#define NSP  32
#define Aa   33          // Lorentz ambient dim
#define APAD 36          // padded to multiple of 4 for WMMA K-steps
#define HA   264         // H*A
#define BT   1024        // B*T
#define EPSF  1e-6f
#define MAXRF 18.0f
#define TINYF 1e-15f
#define TAUF  1.0f
// 1/sqrt(1e-15): the reference's un = sqrt(max(<u,u>_L, TINY)) clamp always
// takes TINY for on-hyperboloid operands (<u,u>_L = -1-3a^2 <= -4), so un is
// a deterministic constant in fp32.
#define INV_UN 3.16227766016838e7f

typedef float v2f __attribute__((ext_vector_type(2)));
typedef float v8f __attribute__((ext_vector_type(8)));

// D(16x16,f32) = A(16x4,f32) * B(4x16,f32) + C   — CDNA5 fp32 matrix pipe
__device__ __forceinline__ v8f wmma4(v2f a, v2f b, v8f c) {
  return __builtin_amdgcn_wmma_f32_16x16x4_f32(
      /*neg_a=*/false, a, /*neg_b=*/false, b,
      /*c_mod=*/(short)0, c, /*reuse_a=*/false, /*reuse_b=*/false);
}

// ---- Tensor Data Mover (CDNA5): 1D linear copy global -> LDS ----
#if defined(__has_builtin)
#if __has_builtin(__builtin_amdgcn_tensor_load_to_lds) && \
    __has_builtin(__builtin_amdgcn_s_wait_tensorcnt)
#define HAVE_TDM 1
#endif
#endif

#ifdef HAVE_TDM
typedef unsigned int u32x4 __attribute__((ext_vector_type(4)));
typedef int          i32x8 __attribute__((ext_vector_type(8)));
typedef int          i32x4 __attribute__((ext_vector_type(4)));

// D# per CDNA5 ISA: 1D tensor of `nelem` 4-byte elements,
// tile == tensor (single contiguous line). Groups 2/3 unused (zero).
// This toolchain exposes the 6-arg builtin form:
//   (u32x4 g0, i32x8 g1, i32x4 g2, i32x4 g3, i32x8 extra, i32 cpol)
__device__ __forceinline__ void tdm_copy_1d_f32(unsigned lds_byte_addr,
                                                const void* gptr,
                                                unsigned nelem) {
  unsigned long long ga = (unsigned long long)gptr;
  u32x4 g0;
  g0[0] = 1u;                                            // count=1 (valid), is_restore=0
  g0[1] = lds_byte_addr;                                 // lds_addr [63:32]
  g0[2] = (unsigned)(ga & 0xffffffffull);                // global_addr [95:64]
  g0[3] = (unsigned)((ga >> 32) & 0x1ffffffull)          // global_addr [120:96]
        | (2u << 30);                                    // type=2 ("image") [127:126]
  i32x8 g1;
  g1[0] = (int)(2u << 16);                               // wg_mask=0, data_size=2 (4B)
  g1[1] = (int)((nelem & 0xffffu) << 16);                // tensor_dim0 [79:48] lo16
  g1[2] = (int)(((nelem >> 16) & 0xffffu))               // tensor_dim0 hi16
        | (int)(1u << 16);                               // tensor_dim1 = 1 [111:80] lo16
  g1[3] = (int)((nelem & 0xffffu) << 16);                // tensor_dim1 hi=0, tile_dim0 [127:112]
  g1[4] = 1;                                             // tile_dim1 = 1, tile_dim2 = 0
  g1[5] = (int)nelem;                                    // tensor_dim0_stride [207:160] lo32
  g1[6] = 0;                                             // stride0 hi16, tensor_dim1_stride lo16
  g1[7] = 0;
  i32x4 z4 = {};
  i32x8 z8 = {};
  __builtin_amdgcn_tensor_load_to_lds(g0, g1, z4, z4, z8, 0);
}
#endif

// A-fragment (16x4): lane<16 holds row=lane, K = k,k+1 ; lane>=16 holds K = k+2,k+3
// B-fragment (4x16): lane<16 holds col=lane, K = k,k+1 ; lane>=16 holds K = k+2,k+3
// C/D (16x16): vgpr r, lane l -> row = r + 8*(l>=16), col = l&15

// ============================================================
// Kernel 1: fused QKV lift GEMM  Z3(1024x768) = x(1024x512) @ [Wq|Wk|Wv]^T + b
// grid (64, 12), block 128 (4 waves); each wave: one 16x16 tile column group
// ============================================================
__global__ void __launch_bounds__(128)
k_lift_gemm(const float* __restrict__ x,
            const float* __restrict__ Wq, const float* __restrict__ bq,
            const float* __restrict__ Wk, const float* __restrict__ bk,
            const float* __restrict__ Wv, const float* __restrict__ bv,
            float* __restrict__ Z3) {
  __shared__ float xs[16 * 512];
  const int m0 = blockIdx.x * 16;
  const int tid = threadIdx.x, lane = tid & 31, wave = tid >> 5;

  // stage 16 rows of x into LDS (vectorized)
  const float4* src = (const float4*)(x + m0 * 512);
  float4* dst = (float4*)xs;
  for (int v = tid; v < 16 * 512 / 4; v += 128) dst[v] = src[v];
  __syncthreads();

  const int n0 = blockIdx.y * 64 + wave * 16;
  const int nglob = n0 + (lane & 15);
  const float* Wsel; const float* bsel; int nn;
  if (nglob < 256)      { Wsel = Wq; bsel = bq; nn = nglob;       }
  else if (nglob < 512) { Wsel = Wk; bsel = bk; nn = nglob - 256; }
  else                  { Wsel = Wv; bsel = bv; nn = nglob - 512; }
  const float* wrow = Wsel + nn * 512;   // B[k][n] = W[n][k]

  const int koff = (lane >> 4) << 1;
  const int arow = lane & 15;
  v8f acc = {};
  for (int k = 0; k < 512; k += 4) {
    v2f a, b;
    a.x = xs[arow * 512 + k + koff];
    a.y = xs[arow * 512 + k + koff + 1];
    b.x = wrow[k + koff];
    b.y = wrow[k + koff + 1];
    acc = wmma4(a, b, acc);
  }
  const float bias = bsel[nn];
  const int rbase = m0 + ((lane >> 4) << 3);
  const int col = n0 + (lane & 15);
  for (int r = 0; r < 8; ++r)
    Z3[(rbase + r) * 768 + col] = acc[r] + bias;
}

// ============================================================
// Kernel 2: FL (Euclidean -> Lorentz) elementwise; writes (B,H,T,APAD), pad=0
// ============================================================
__global__ void k_fl(const float* __restrict__ Z3, float* __restrict__ Qh,
                     float* __restrict__ Kh, float* __restrict__ Vh) {
  const int idx = blockIdx.x * blockDim.x + threadIdx.x;
  const int per = Bv * Hh * Tv;
  if (idx >= 3 * per) return;
  const int which = idx / per;
  int rem = idx - which * per;
  const int b = rem / (Hh * Tv); rem -= b * (Hh * Tv);
  const int h = rem / Tv;
  const int t = rem - h * Tv;

  const float* z = Z3 + (b * Tv + t) * 768 + which * 256 + h * 32;
  float zz[32]; float r2 = 0.f;
  for (int i = 0; i < 32; ++i) { zz[i] = z[i]; r2 += zz[i] * zz[i]; }
  const float r  = sqrtf(r2);
  const float rs = fminf(fmaxf(r, 1e-12f), MAXRF);
  const float y0 = coshf(rs);
  const float sh = sinhf(rs) / rs;       // sinh(r)*z/r
  float ys[32]; float sp2 = 0.f;
  for (int i = 0; i < 32; ++i) { ys[i] = sh * zz[i]; sp2 += ys[i] * ys[i]; }
  // project_to_hyperboloid
  const float xx  = fmaxf(fabsf(-y0 * y0 + sp2), TINYF);
  const float inv = 1.0f / sqrtf(xx);
  float* out = (which == 0 ? Qh : which == 1 ? Kh : Vh)
             + ((b * Hh + h) * Tv + t) * APAD;
  out[0] = fabsf(y0 * inv);
  for (int i = 0; i < 32; ++i) out[1 + i] = ys[i] * inv;
  out[33] = 0.f; out[34] = 0.f; out[35] = 0.f;
}

// ============================================================
// Kernel 3: fused attention + Karcher step per (b,h) / 16-row tile.
// K,V for the whole (b,h) live in LDS (2*73.7KB), DMA'd by the Tensor Data
// Mover; score matrices never hit HBM.
//   ip    = Q~ * K^T   (WMMA, K-dim = APAD)      Q~ = metric-flipped Q
//   alpha = Q~ * V^T   (WMMA)
//   softmax(-acosh^2(ip)) ; w = attn*acosh(alpha) ; s_i = sum w*alpha
//   Uv    = w(16x512) * V(512x36)  (WMMA, K-dim = 512)
//   exp_map / project / origin log_map  -> Ztan (B,T,H*A)
// ============================================================
__global__ void __launch_bounds__(128)
k_attn(const float* __restrict__ Qh, const float* __restrict__ Kh,
       const float* __restrict__ Vh, float* __restrict__ Ztan) {
  __shared__ float smem[2 * Tv * APAD + 2 * 16 * APAD + 2 * 16 * Tv
                        + 16 * 48 + 128 + 48];     // 55344 f = 221.4 KB (<320KB WGP LDS)
  float* Ks  = smem;                  // 512*36   (LDS byte offset 0)
  float* Vs  = Ks + Tv * APAD;        // 512*36   (LDS byte offset 73728)
  float* Qt  = Vs + Tv * APAD;        // 16*36  (metric-flipped Q)
  float* Yt  = Qt + 16 * APAD;        // 16*36  (raw Q = Karcher init Y)
  float* Ssc = Yt + 16 * APAD;        // 16*512 scores, later reused for weights w
  float* Sal = Ssc + 16 * Tv;         // 16*512 alpha
  float* Uvs = Sal + 16 * Tv;         // 16*48
  float* red = Uvs + 16 * 48;         // 16*8
  float* rowmax = red + 128;          // 16
  float* rowsum = rowmax + 16;        // 16
  float* rowS   = rowsum + 16;        // 16

  const int bh = blockIdx.x >> 5;
  const int i0 = (blockIdx.x & 31) << 4;
  const int b  = bh >> 3, h = bh & 7;
  const int tid = threadIdx.x, lane = tid & 31, wave = tid >> 5;

#ifdef HAVE_TDM
  // Wave 0 issues two TDM descriptors (one per slab); the DMA runs while the
  // whole block stages the Q tile below. smem is this kernel's only LDS
  // object, allocated at LDS base, so sub-array offsets are absolute.
  if (wave == 0) {
    tdm_copy_1d_f32(0u,                    Kh + bh * Tv * APAD, Tv * APAD);
    tdm_copy_1d_f32((unsigned)(Tv * APAD * sizeof(float)),
                                           Vh + bh * Tv * APAD, Tv * APAD);
  }
#else
  {
    const float4* ksrc = (const float4*)(Kh + bh * Tv * APAD);
    const float4* vsrc = (const float4*)(Vh + bh * Tv * APAD);
    float4* kdst = (float4*)Ks; float4* vdst = (float4*)Vs;
    for (int v = tid; v < Tv * APAD / 4; v += 128) { kdst[v] = ksrc[v]; vdst[v] = vsrc[v]; }
  }
#endif

  // stage Q tile; build metric-flipped copy (ip = Q~ . K , alpha = Q~ . V)
  for (int v = tid; v < 16 * APAD; v += 128) {
    const int a = v % APAD;
    const float q = Qh[bh * Tv * APAD + i0 * APAD + v];
    Yt[v] = q;
    Qt[v] = (a == 0) ? q : -q;
  }
#ifdef HAVE_TDM
  if (wave == 0) __builtin_amdgcn_s_wait_tensorcnt(0);  // issuing wave drains TENSORcnt
#endif
  __syncthreads();

  const int koff = (lane >> 4) << 1;
  const int arow = lane & 15;

  // ---- score GEMMs over all 32 j-tiles (8 per wave) ----
  for (int jt = wave; jt < 32; jt += 4) {
    const int j0 = jt << 4;
    const int jn = j0 + (lane & 15);
    v8f cip = {}, cal = {};
    for (int k = 0; k < APAD; k += 4) {
      v2f a;  a.x  = Qt[arow * APAD + k + koff]; a.y  = Qt[arow * APAD + k + koff + 1];
      v2f bk_; bk_.x = Ks[jn * APAD + k + koff]; bk_.y = Ks[jn * APAD + k + koff + 1];
      v2f bv_; bv_.x = Vs[jn * APAD + k + koff]; bv_.y = Vs[jn * APAD + k + koff + 1];
      cip = wmma4(a, bk_, cip);
      cal = wmma4(a, bv_, cal);
    }
    const int ib = (lane >> 4) << 3;
    for (int r = 0; r < 8; ++r) {
      const float d = acoshf(fmaxf(cip[r], 1.0f + EPSF));
      Ssc[(ib + r) * Tv + jn] = -TAUF * d * d;
      Sal[(ib + r) * Tv + jn] = cal[r];
    }
  }
  __syncthreads();

  // ---- softmax + weight build: 8 threads per row ----
  const int row = tid >> 3, sub = tid & 7;
  {
    float m = -3.4e38f;
    for (int j = sub * 64; j < sub * 64 + 64; ++j) m = fmaxf(m, Ssc[row * Tv + j]);
    red[row * 8 + sub] = m;
  }
  __syncthreads();
  if (tid < 16) {
    float m = red[tid * 8];
    for (int s = 1; s < 8; ++s) m = fmaxf(m, red[tid * 8 + s]);
    rowmax[tid] = m;
  }
  __syncthreads();
  {
    const float m = rowmax[row];
    float s = 0.f;
    for (int j = sub * 64; j < sub * 64 + 64; ++j) s += expf(Ssc[row * Tv + j] - m);
    red[row * 8 + sub] = s;
  }
  __syncthreads();
  if (tid < 16) {
    float s = 0.f;
    for (int k = 0; k < 8; ++k) s += red[tid * 8 + k];
    rowsum[tid] = s;
  }
  __syncthreads();
  {
    const float m = rowmax[row], inv = 1.0f / rowsum[row];
    float sacc = 0.f;
    for (int j = sub * 64; j < sub * 64 + 64; ++j) {
      const float attn = expf(Ssc[row * Tv + j] - m) * inv;
      const float al   = Sal[row * Tv + j];
      const float dv   = acoshf(fmaxf(al, 1.0f + EPSF));
      const float w    = attn * dv;
      Ssc[row * Tv + j] = w;               // reuse score slab as weight matrix
      sacc += w * al;
    }
    red[row * 8 + sub] = sacc;
  }
  __syncthreads();
  if (tid < 16) {
    float s = 0.f;
    for (int k = 0; k < 8; ++k) s += red[tid * 8 + k];
    rowS[tid] = s;
  }
  __syncthreads();

  // ---- aggregation GEMM: Uv(16x48) = w(16x512) * V(512x36) ----
  if (wave < 3) {                          // wave-uniform: EXEC all-1s inside
    const int n0 = wave << 4;
    const int n  = n0 + (lane & 15);
    v8f acc = {};
    for (int k = 0; k < Tv; k += 4) {
      v2f a;  a.x = Ssc[arow * Tv + k + koff]; a.y = Ssc[arow * Tv + k + koff + 1];
      v2f bb;
      bb.x = (n < APAD) ? Vs[(k + koff)     * APAD + n] : 0.f;
      bb.y = (n < APAD) ? Vs[(k + koff + 1) * APAD + n] : 0.f;
      acc = wmma4(a, bb, acc);
    }
    const int ib = (lane >> 4) << 3;
    for (int r = 0; r < 8; ++r) Uvs[(ib + r) * 48 + n] = acc[r];
  }
  __syncthreads();

  // ---- exp_map at Y, project, origin log_map ----
  if (tid < 16) {
    const int i = tid;
    const float s = rowS[i];
    float u[Aa], yv[Aa];
    for (int a = 0; a < Aa; ++a)
      u[a] = (Uvs[i * 48 + a] + s * Yt[i * APAD + a]) * INV_UN;
    float uu = -u[0] * u[0];
    for (int a = 1; a < Aa; ++a) uu += u[a] * u[a];
    const float vn = sqrtf(fmaxf(uu, TINYF));
    const float ch = coshf(vn), sh = sinhf(vn) / fmaxf(vn, TINYF);
    for (int a = 0; a < Aa; ++a) yv[a] = ch * Yt[i * APAD + a] + sh * u[a];
    float yy = -yv[0] * yv[0];
    for (int a = 1; a < Aa; ++a) yy += yv[a] * yv[a];
    const float inv = 1.0f / sqrtf(fmaxf(fabsf(yy), TINYF));
    const float Y0  = fabsf(yv[0] * inv);
    // log_map at Lorentz origin: u0 = (2*Y0, Y_spatial), un clamps to sqrt(TINY)
    const float d0  = acoshf(fmaxf(Y0, 1.0f + EPSF));
    const float sc0 = d0 * INV_UN;
    float* zt = Ztan + (b * Tv + i0 + i) * HA + h * Aa;
    zt[0] = sc0 * 2.0f * Y0;
    for (int a = 1; a < Aa; ++a) zt[a] = sc0 * (yv[a] * inv);
  }
}

// ============================================================
// Kernel 4: output GEMM  out(1024x512) = Ztan(1024x264) @ Wo^T + bo
// ============================================================
__global__ void __launch_bounds__(128)
k_out_gemm(const float* __restrict__ Zt, const float* __restrict__ Wo,
           const float* __restrict__ bo, float* __restrict__ out) {
  __shared__ float zs[16 * HA];
  const int m0 = blockIdx.x * 16;
  const int tid = threadIdx.x, lane = tid & 31, wave = tid >> 5;

  const float4* src = (const float4*)(Zt + m0 * HA);
  float4* dst = (float4*)zs;
  for (int v = tid; v < 16 * HA / 4; v += 128) dst[v] = src[v];
  __syncthreads();

  const int n0 = blockIdx.y * 64 + wave * 16;
  const int nglob = n0 + (lane & 15);
  const float* wrow = Wo + nglob * HA;   // B[k][n] = Wo[n][k]
  const int koff = (lane >> 4) << 1;
  const int arow = lane & 15;
  v8f acc = {};
  for (int k = 0; k < HA; k += 4) {
    v2f a, b;
    a.x = zs[arow * HA + k + koff];
    a.y = zs[arow * HA + k + koff + 1];
    b.x = wrow[k + koff];
    b.y = wrow[k + koff + 1];
    acc = wmma4(a, b, acc);
  }
  const float bias = bo[nglob];
  const int rbase = m0 + ((lane >> 4) << 3);
  const int col = n0 + (lane & 15);
  for (int r = 0; r < 8; ++r)
    out[(rbase + r) * Dm + col] = acc[r] + bias;
}

// ============================================================
extern "C" void kernel_launch(void* const* d_in, const int* in_sizes, int n_in,
                              void* d_out, int out_size, void* d_ws, size_t ws_size,
                              hipStream_t stream) {
  const float* x  = (const float*)d_in[0];
  const float* Wq = (const float*)d_in[1];
  const float* bq = (const float*)d_in[2];
  const float* Wk = (const float*)d_in[3];
  const float* bk = (const float*)d_in[4];
  const float* Wv = (const float*)d_in[5];
  const float* bv = (const float*)d_in[6];
  const float* Wo = (const float*)d_in[7];
  const float* bo = (const float*)d_in[8];
  float* out = (float*)d_out;

  // workspace layout (floats): ~7.8 MB total
  float* ws = (float*)d_ws;
  float* Z3 = ws;                          // 1024*768
  float* Qh = Z3 + BT * 768;               // B*H*T*APAD = 294912 each
  float* Kh = Qh + Bv * Hh * Tv * APAD;
  float* Vh = Kh + Bv * Hh * Tv * APAD;
  float* Zt = Vh + Bv * Hh * Tv * APAD;    // 1024*264

  k_lift_gemm<<<dim3(BT / 16, 768 / 64), 128, 0, stream>>>(x, Wq, bq, Wk, bk, Wv, bv, Z3);
  k_fl<<<(3 * Bv * Hh * Tv + 255) / 256, 256, 0, stream>>>(Z3, Qh, Kh, Vh);
  k_attn<<<dim3(Bv * Hh * 32), 128, 0, stream>>>(Qh, Kh, Vh, Zt);
  k_out_gemm<<<dim3(BT / 16, Dm / 64), 128, 0, stream>>>(Zt, Wo, bo, out);
}